// Attn_90589450207860
// MI455X (gfx1250) — compile-verified
//
#include <hip/hip_runtime.h>
#include <hip/hip_bf16.h>

typedef __attribute__((ext_vector_type(2))) float v2f;
typedef __attribute__((ext_vector_type(4))) float f4;
typedef __attribute__((ext_vector_type(8))) float v8f;

#define HDIM 1024
#define TDIM 4096
#define BDIM 16

// ---------------------------------------------------------------------------
// Phase 1: qT[b][h] = sum_d W[d][h] * dec[d][b]      (q = W^T @ dec_hidden)
// One wave (32 lanes) per 16-wide h-tile. M=16 (h), N=16 (b), K=1024 (d),
// accumulated with V_WMMA_F32_16X16X4_F32 in K-steps of 4.
//
// ISA 32-bit A-matrix 16x4 layout: lanes 0-15 -> M=lane, VGPR0=K0/VGPR1=K1;
// lanes 16-31 -> M=lane-16, VGPR0=K2/VGPR1=K3. B (4x16) mirrors this with
// N across lanes. C/D: VGPR r -> row M=r (lanes 0-15) / M=r+8 (lanes 16-31).
// ---------------------------------------------------------------------------
__global__ __launch_bounds__(32) void qproj_wmma(const float* __restrict__ W,
                                                 const float* __restrict__ dec,
                                                 float* __restrict__ qT) {
  const int h0    = blockIdx.x * 16;
  const int lane  = threadIdx.x;
  const int m     = lane & 15;          // M (A) / N (B) index for this lane
  const int khalf = (lane >> 4) * 2;    // 0 for lanes 0-15, 2 for lanes 16-31

#if __has_builtin(__builtin_amdgcn_wmma_f32_16x16x4_f32)
  v8f acc = {};
  for (int d0 = 0; d0 < HDIM; d0 += 4) {
    v2f a, b;
    // A[i,k] = W^T[h0+i, d0+k] = W[d0+k, h0+i]
    a[0] = W[(size_t)(d0 + khalf)     * HDIM + h0 + m];
    a[1] = W[(size_t)(d0 + khalf + 1) * HDIM + h0 + m];
    // B[k,n] = dec[d0+k, n]
    b[0] = dec[(d0 + khalf)     * BDIM + m];
    b[1] = dec[(d0 + khalf + 1) * BDIM + m];
    acc = __builtin_amdgcn_wmma_f32_16x16x4_f32(
        /*neg_a=*/false, a, /*neg_b=*/false, b,
        /*c_mod=*/(short)0, acc, /*reuse_a=*/false, /*reuse_b=*/false);
  }
#else
  // Scalar fallback producing the identical D-tile register layout.
  v8f acc = {};
  const int mb_f = (lane < 16) ? 0 : 8;
  for (int d = 0; d < HDIM; ++d) {
    const float db = dec[d * BDIM + m];
    #pragma unroll
    for (int r = 0; r < 8; ++r)
      acc[r] = fmaf(W[(size_t)d * HDIM + h0 + mb_f + r], db, acc[r]);
  }
#endif

  // Store D: row M = mb + r, col N = m  ->  qT[b][h] row-major per b.
  const int mb = (lane < 16) ? 0 : 8;
  #pragma unroll
  for (int r = 0; r < 8; ++r)
    qT[(size_t)m * HDIM + h0 + mb + r] = acc[r];
}

// ---------------------------------------------------------------------------
// Phase 2 (bandwidth-dominant): scores[b,t] = sum_h enc[t,b,h] * qT[b,h].
// Streams 268 MB of encoder_outputs once -> ~11.5 us at 23.3 TB/s.
// Block = 256 threads = 8 waves; block covers (b, 64 consecutive t).
// Each wave caches q[:,b] in 8 float4 regs/lane and does 8 t-rows.
// enc loads: lane-consecutive float4 = 512B/inst, NT hint (read-once).
// ---------------------------------------------------------------------------
__global__ __launch_bounds__(256) void scores_kernel(const float* __restrict__ enc,
                                                     const float* __restrict__ qT,
                                                     float* __restrict__ out) {
  const int b    = blockIdx.y;
  const int lane = threadIdx.x & 31;
  const int wave = threadIdx.x >> 5;

  const f4* qv4 = (const f4*)(qT + (size_t)b * HDIM);
  f4 q[8];
  #pragma unroll
  for (int j = 0; j < 8; ++j) q[j] = qv4[j * 32 + lane];

  const int tbase = blockIdx.x * 64 + wave * 8;
  #pragma unroll
  for (int i = 0; i < 8; ++i) {
    const int t = tbase + i;
    const f4* ep = (const f4*)(enc + ((size_t)t * BDIM + b) * HDIM);
    float acc = 0.f;
    #pragma unroll
    for (int j = 0; j < 8; ++j) {
      const f4 e = __builtin_nontemporal_load(&ep[j * 32 + lane]);
      acc = fmaf(e[0], q[j][0], acc);
      acc = fmaf(e[1], q[j][1], acc);
      acc = fmaf(e[2], q[j][2], acc);
      acc = fmaf(e[3], q[j][3], acc);
    }
    #pragma unroll
    for (int off = 16; off > 0; off >>= 1)
      acc += __shfl_xor(acc, off, 32);
    if (lane == 0) out[(size_t)b * TDIM + t] = acc;
  }
}

// ---------------------------------------------------------------------------
// Phase 3: in-place softmax over t for each of the 16 rows (max-shifted).
// ---------------------------------------------------------------------------
__global__ __launch_bounds__(256) void softmax_rows(float* __restrict__ out) {
  __shared__ float red[256];
  const int b = blockIdx.x;
  float* row = out + (size_t)b * TDIM;
  const int tid = threadIdx.x;

  float m = -3.402823466e+38f;
  for (int t = tid; t < TDIM; t += 256) m = fmaxf(m, row[t]);
  red[tid] = m; __syncthreads();
  for (int s = 128; s > 0; s >>= 1) {
    if (tid < s) red[tid] = fmaxf(red[tid], red[tid + s]);
    __syncthreads();
  }
  m = red[0];
  __syncthreads();

  float sum = 0.f;
  for (int t = tid; t < TDIM; t += 256) sum += expf(row[t] - m);
  red[tid] = sum; __syncthreads();
  for (int s = 128; s > 0; s >>= 1) {
    if (tid < s) red[tid] += red[tid + s];
    __syncthreads();
  }
  const float inv = 1.0f / red[0];

  for (int t = tid; t < TDIM; t += 256) row[t] = expf(row[t] - m) * inv;
}

// ---------------------------------------------------------------------------
// Inputs (setup_inputs order): d_in[0]=dec_hidden [H,B] f32,
// d_in[1]=encoder_outputs [T,B,H] f32, d_in[2]=W [H,H] f32.
// d_out: [B,T] f32. d_ws: qT[B][H] = 64 KB.
// ---------------------------------------------------------------------------
extern "C" void kernel_launch(void* const* d_in, const int* in_sizes, int n_in,
                              void* d_out, int out_size, void* d_ws, size_t ws_size,
                              hipStream_t stream) {
  const float* dec = (const float*)d_in[0];
  const float* enc = (const float*)d_in[1];
  const float* W   = (const float*)d_in[2];
  float* out = (float*)d_out;
  float* qT  = (float*)d_ws;   // B*H floats = 64 KB

  qproj_wmma<<<HDIM / 16, 32, 0, stream>>>(W, dec, qT);
  scores_kernel<<<dim3(TDIM / 64, BDIM), 256, 0, stream>>>(enc, qT, out);
  softmax_rows<<<BDIM, 256, 0, stream>>>(out);
}